// ErdosLoss_80900003988334
// MI455X (gfx1250) — compile-verified
//
#include <hip/hip_runtime.h>
#include <hip/hip_bf16.h>

#define N_NODES  3072
#define N_EDGES  6144
// PENALTY_SCALE = 16*200*3 = 9600

typedef __attribute__((ext_vector_type(2))) float v2f;
typedef __attribute__((ext_vector_type(8))) float v8f;

// Exact f32 wave32 reduction via V_WMMA_F32_16X16X4_F32.
// A holds one partial per lane (other A slot = 0), B = all ones, C = 0.
// D[m,n] = rowsum_m(A) for every column n, so
//   sum of my 8 D regs (M = 0..7 or 8..15 at column n=lane%16)
// + same from lane^16  == total of all 64 A slots == wave sum.
__device__ __forceinline__ float wave_reduce_wmma(float v) {
    v2f a; a.x = v;    a.y = 0.0f;
    v2f b; b.x = 1.0f; b.y = 1.0f;
    v8f c = {};
    c = __builtin_amdgcn_wmma_f32_16x16x4_f32(
            /*neg_a=*/false, a, /*neg_b=*/false, b,
            /*c_mod=*/(short)0, c, /*reuse_a=*/false, /*reuse_b=*/false);
    float s = c[0] + c[1] + c[2] + c[3] + c[4] + c[5] + c[6] + c[7];
    s += __shfl_xor(s, 16, 32);
    return s;   // all lanes hold the wave total
}

__global__ __launch_bounds__(1024)
void erdos_fused(const float* __restrict__ x,
                 const float* __restrict__ p,        // edge_feature (E,1)
                 const float* __restrict__ w,        // w_proxy (1,1)
                 const int*   __restrict__ ei,       // edge_index (2,E)
                 const int*   __restrict__ batch,    // sorted (N,)
                 float*       __restrict__ out)
{
    __shared__ float s_arr[N_NODES];    // s_n = sum of p_e over incident edges
    __shared__ float ls_arr[N_NODES];   // log_score accumulators
    __shared__ float red0[32], red1[32], red2[32], red3[32];

    const int tid = threadIdx.x;

    for (int i = tid; i < N_NODES; i += 1024) { s_arr[i] = 0.0f; ls_arr[i] = 0.0f; }
    __syncthreads();

    const int* src = ei;
    const int* dst = ei + N_EDGES;

    // ---- edge phase: scatter into LDS ----
    float diag = 0.0f;                          // sum p_e^2 * d_e
    for (int e = tid; e < N_EDGES; e += 1024) { // 6 edges per thread, no divergence
        float pe = p[e];
        int s = src[e], d = dst[e];
        float msg = logf(1.0f - pe + 1e-6f);
        atomicAdd(&ls_arr[d], msg);
        atomicAdd(&s_arr[s], pe);
        if (d != s) { atomicAdd(&s_arr[d], pe); diag += 2.0f * pe * pe; }
        else        { diag += pe * pe; }
    }

    // ---- x sum (independent of LDS scatter) ----
    float sumx = 0.0f;
    for (int n = tid; n < N_NODES; n += 1024) sumx += x[n];

    __syncthreads();

    // ---- node phase: gather from LDS ----
    float sumexp = 0.0f, sums2 = 0.0f;
    for (int n = tid; n < N_NODES; n += 1024) {
        sumexp += expf(ls_arr[n]);              // no-incoming-edge nodes give exp(0)=1
        float sn = s_arr[n];
        sums2 = fmaf(sn, sn, sums2);
    }

    // ---- per-wave reduction on the matrix core (EXEC all ones here) ----
    float wsums2 = wave_reduce_wmma(sums2);
    float wsume  = wave_reduce_wmma(sumexp);
    float wsumx  = wave_reduce_wmma(sumx);
    float wdiag  = wave_reduce_wmma(diag);

    const int wave = tid >> 5, lane = tid & 31;
    if (lane == 0) { red0[wave] = wsums2; red1[wave] = wsume;
                     red2[wave] = wsumx;  red3[wave] = wdiag; }
    __syncthreads();

    if (tid == 0) {
        float S2 = 0.0f, SE = 0.0f, SX = 0.0f, DG = 0.0f;
        for (int i = 0; i < 32; ++i) {
            S2 += red0[i]; SE += red1[i]; SX += red2[i]; DG += red3[i];
        }
        float ng    = (float)batch[N_NODES - 1] + 1.0f;   // batch is sorted -> max is last
        float term1 = w[0] * SX * (1.0f / 32.0f);         // mean(loss_1)
        float term2 = SE * (9600.0f / (float)N_NODES);    // mean(loss_2)
        float loss3 = 0.5f * (S2 - DG);                   // p M p^T, strict upper tri
        out[0] = term1 + term2 + (loss3 / ng) * 200.0f;
    }
}

extern "C" void kernel_launch(void* const* d_in, const int* in_sizes, int n_in,
                              void* d_out, int out_size, void* d_ws, size_t ws_size,
                              hipStream_t stream) {
    const float* x     = (const float*)d_in[0];   // (3072,)
    const float* p     = (const float*)d_in[1];   // (6144,1)
    const float* w     = (const float*)d_in[2];   // (1,1)
    const int*   ei    = (const int*)  d_in[3];   // (2,6144)
    const int*   batch = (const int*)  d_in[4];   // (3072,)
    float*       out   = (float*)d_out;

    erdos_fused<<<1, 1024, 0, stream>>>(x, p, w, ei, batch, out);
}